// GraphLoss_52037823758709
// MI455X (gfx1250) — compile-verified
//
#include <hip/hip_runtime.h>

#define W 128
#define LM1 2000                       // number of scan steps (L-1)
#define LAYER_ELEMS (W * W)            // 16384 floats = 64 KB per layer
#define NBUF 4                         // LDS ring depth (4 * 64KB = 256KB of 320KB)
#define NB_GOLD 2048                   // gold-score partial blocks

typedef __attribute__((ext_vector_type(2)))  float        v2f;
typedef __attribute__((ext_vector_type(8)))  float        v8f;
typedef __attribute__((ext_vector_type(4)))  unsigned int u32x4;
typedef __attribute__((ext_vector_type(8)))  int          i32x8;
typedef __attribute__((ext_vector_type(4)))  int          i32x4;

#if __has_builtin(__builtin_amdgcn_wmma_f32_16x16x4_f32)
#define USE_WMMA_F32 1
#else
#define USE_WMMA_F32 0
#endif

#if __has_builtin(__builtin_amdgcn_tensor_load_to_lds) && __has_builtin(__builtin_amdgcn_s_wait_tensorcnt)
#define USE_TDM 1
#else
#define USE_TDM 0
#endif

__device__ __forceinline__ unsigned lds_byte_off(const void* p) {
    // Flat->LDS mapping truncates to addr[31:0]; group segment starts at 0.
    return (unsigned)(unsigned long long)(uintptr_t)p;
}

#if USE_TDM
// Issue one TDM descriptor moving a full 64KB layer (1-D tile of 16384 f32)
// from global memory into LDS. Tracked with TENSORcnt.
__device__ __forceinline__ void tdm_load_layer(const float* gsrc, unsigned lds_off) {
    unsigned long long ga = (unsigned long long)(uintptr_t)gsrc;
    u32x4 g0;
    g0[0] = 1u;                                        // count=1, user descriptor
    g0[1] = lds_off;                                   // lds_addr (bytes)
    g0[2] = (unsigned)(ga & 0xFFFFFFFFull);            // global_addr[31:0]
    g0[3] = (unsigned)((ga >> 32) & 0x1FFFFFFull)      // global_addr[56:32]
          | (2u << 30);                                // type=2 ("image")
    i32x8 g1 = {};
    g1[0] = (int)(2u << 16);                           // workgroup_mask=0, data_size=4B
    g1[1] = (int)((LAYER_ELEMS & 0xFFFFu) << 16);      // tensor_dim0[15:0] in [31:16]
    g1[2] = (int)(((unsigned)LAYER_ELEMS >> 16) | (1u << 16)); // td0 hi, tensor_dim1=1
    g1[3] = (int)((LAYER_ELEMS & 0xFFFFu) << 16);      // tile_dim0 = 16384
    g1[4] = 0;                                         // tile_dim1=0 (1-D), tile_dim2=0
    g1[5] = LAYER_ELEMS;                               // tensor_dim0_stride lo32
    g1[6] = 0;
    g1[7] = 0;
    i32x4 g2 = {};
    i32x4 g3 = {};
    i32x8 g4 = {};                                     // extra group (6-arg toolchain)
    __builtin_amdgcn_tensor_load_to_lds(g0, g1, g2, g3, g4, 0);
}
#endif

// ---------------------------------------------------------------------------
// Kernel 1: gold-score block partials. Fixed assignment + fixed reduction order
// -> bitwise deterministic (no float atomics).
// ---------------------------------------------------------------------------
__global__ void __launch_bounds__(256)
gold_partial_kernel(const float* __restrict__ w, const int* __restrict__ mask,
                    float* __restrict__ partials, int n) {
    __shared__ float red[256];
    float s = 0.f;
    int stride = (int)(gridDim.x * blockDim.x);
    for (int idx = (int)(blockIdx.x * blockDim.x + threadIdx.x); idx < n; idx += stride)
        s = fmaf(w[idx], (float)mask[idx], s);
    red[threadIdx.x] = s;
    __syncthreads();
    for (int off = 128; off > 0; off >>= 1) {
        if ((int)threadIdx.x < off) red[threadIdx.x] += red[threadIdx.x + off];
        __syncthreads();
    }
    if (threadIdx.x == 0) partials[blockIdx.x] = red[0];
}

// ---------------------------------------------------------------------------
// Kernel 2: persistent single-WGP CRF forward scan.
//   8 waves x 32 lanes; wave w owns columns [16w, 16w+16).
//   Per step: M = max_i e[i]; B[k][j] = exp(e_i - M - w_ij) fed into
//   V_WMMA_F32_16X16X4_F32 with A = ones => D[*][j] = sum_i B -> s_j.
//   e'[j] = M + log(s_j).  TDM prefetches layers 3 deep into a 4-slot ring.
// ---------------------------------------------------------------------------
__global__ void __launch_bounds__(256)
crf_forward_kernel(const float* __restrict__ weight,
                   const float* __restrict__ gold_partials,
                   float* __restrict__ out) {
    extern __shared__ float smem[];
    float* wring = smem;                       // NBUF * 16384 floats
    float* eb0   = smem + NBUF * LAYER_ELEMS;  // 128
    float* eb1   = eb0 + W;                    // 128

    const int tid  = (int)threadIdx.x;
    const int lane = tid & 31;
    const int wave = tid >> 5;
    const int half = lane >> 4;                // B-layout lane half -> K offset
    const int j    = (wave << 4) + (lane & 15);

    const float* wtrans = weight + W;

    if (tid < W) eb0[tid] = -weight[tid];      // e0 = -w_start

#if USE_TDM
    if (wave == 0) {
        for (int t = 0; t < NBUF - 1; ++t)
            tdm_load_layer(wtrans + (size_t)t * LAYER_ELEMS,
                           lds_byte_off(wring + (t % NBUF) * LAYER_ELEMS));
    }
#endif

    int p = 0;
    for (int t = 0; t < LM1; ++t) {
#if USE_TDM
        if (wave == 0) __builtin_amdgcn_s_wait_tensorcnt(NBUF - 2); // layer t done
        __syncthreads();   // layer t visible; everyone done with slot (t-1)%NBUF
        if (wave == 0 && (t + NBUF - 1) < LM1)
            tdm_load_layer(wtrans + (size_t)(t + NBUF - 1) * LAYER_ELEMS,
                           lds_byte_off(wring + ((t + NBUF - 1) % NBUF) * LAYER_ELEMS));
        const float* wb = wring + (t % NBUF) * LAYER_ELEMS;
#else
        __syncthreads();
        {   // synchronous fallback copy, single slot
            const float4* src = (const float4*)(wtrans + (size_t)t * LAYER_ELEMS);
            float4*       dst = (float4*)wring;
            for (int idx = tid; idx < LAYER_ELEMS / 4; idx += 256) dst[idx] = src[idx];
        }
        __syncthreads();
        const float* wb = wring;
#endif
        const float* ec = p ? eb1 : eb0;
        float*       en = p ? eb0 : eb1;

        // global shift M = max_i e[i], redundantly per wave (no extra barrier)
        float m = fmaxf(fmaxf(ec[lane], ec[lane + 32]),
                        fmaxf(ec[lane + 64], ec[lane + 96]));
        for (int d = 16; d; d >>= 1) m = fmaxf(m, __shfl_xor(m, d));

#if USE_WMMA_F32
        v8f acc0 = {};
        v8f acc1 = {};
        const v2f ones = {1.0f, 1.0f};
#pragma unroll
        for (int k0 = 0; k0 < 32; k0 += 2) {
            int i0 = (k0 << 2) + (half << 1);
            v2f b;
            b.x = __expf(ec[i0]     - m - wb[i0 * W + j]);
            b.y = __expf(ec[i0 + 1] - m - wb[(i0 + 1) * W + j]);
            acc0 = __builtin_amdgcn_wmma_f32_16x16x4_f32(
                false, ones, false, b, (short)0, acc0, false, false);
            int i1 = ((k0 + 1) << 2) + (half << 1);
            v2f c;
            c.x = __expf(ec[i1]     - m - wb[i1 * W + j]);
            c.y = __expf(ec[i1 + 1] - m - wb[(i1 + 1) * W + j]);
            acc1 = __builtin_amdgcn_wmma_f32_16x16x4_f32(
                false, ones, false, c, (short)0, acc1, false, false);
        }
        // all D rows identical (A = ones): acc[0] = s_{j} on every lane
        float s = acc0[0] + acc1[0];
#else
        // VALU fallback: lane sums its 64 i-values; combine the two lane-halves
        float s = 0.f;
#pragma unroll
        for (int k0 = 0; k0 < 32; ++k0) {
            int i0 = (k0 << 2) + (half << 1);
            s += __expf(ec[i0]     - m - wb[i0 * W + j]);
            s += __expf(ec[i0 + 1] - m - wb[(i0 + 1) * W + j]);
        }
        s += __shfl_xor(s, 16);
#endif
        if (lane < 16) en[j] = m + __logf(s);
        p ^= 1;
    }

    __syncthreads();
    if (wave == 0) {
        const float* ef   = p ? eb1 : eb0;
        const float* wend = weight + W + (size_t)LM1 * LAYER_ELEMS;
        float x0 = ef[lane]      - wend[lane];
        float x1 = ef[lane + 32] - wend[lane + 32];
        float x2 = ef[lane + 64] - wend[lane + 64];
        float x3 = ef[lane + 96] - wend[lane + 96];
        float mx = fmaxf(fmaxf(x0, x1), fmaxf(x2, x3));
        for (int d = 16; d; d >>= 1) mx = fmaxf(mx, __shfl_xor(mx, d));
        float s = __expf(x0 - mx) + __expf(x1 - mx) + __expf(x2 - mx) + __expf(x3 - mx);
        for (int d = 16; d; d >>= 1) s += __shfl_xor(s, d);
        float fwd = mx + __logf(s);

        float g = 0.f;
        for (int idx = lane; idx < NB_GOLD; idx += 32) g += gold_partials[idx];
        for (int d = 16; d; d >>= 1) g += __shfl_xor(g, d);

        if (lane == 0) out[0] = g + fwd;
    }
}

extern "C" void kernel_launch(void* const* d_in, const int* in_sizes, int n_in,
                              void* d_out, int out_size, void* d_ws, size_t ws_size,
                              hipStream_t stream) {
    const float* weight = (const float*)d_in[0];
    const int*   mask   = (const int*)d_in[1];
    float*       out    = (float*)d_out;
    float*       parts  = (float*)d_ws;     // NB_GOLD partial sums
    int n = in_sizes[0];                    // N_EDGES = 32,768,256

    gold_partial_kernel<<<NB_GOLD, 256, 0, stream>>>(weight, mask, parts, n);

    size_t smem = (size_t)NBUF * LAYER_ELEMS * sizeof(float) + 2 * W * sizeof(float);
    (void)hipFuncSetAttribute((const void*)crf_forward_kernel,
                              hipFuncAttributeMaxDynamicSharedMemorySize, (int)smem);
    crf_forward_kernel<<<1, 256, smem, stream>>>(weight, parts, out);
}